// Model_73023033966805
// MI455X (gfx1250) — compile-verified
//
#include <hip/hip_runtime.h>
#include <hip/hip_bf16.h>
#include <math.h>

// ---------------- Instant-NGP constants (compile-time) ----------------
// L=10 levels, base res 16, C=4 feats, hashmap 2^21, align_corners=False.
// Level table (res=16<<l, side=res+1):
//  l : params(padded)  offset      hashed
//  0 : 4920            0           no
//  1 : 35944           4920        no
//  2 : 274632          40864       no
//  3 : 2097152         315496      yes   (2^21, power of two -> mask)
//  4..9: 2097152 each, offsets +2097152
#define NGP_L 10
#define NGP_C 4
#define HASH_MASK 2097151u       // 2^21 - 1
#define P1 2654435761u
#define P2 805459861u

__device__ __constant__ int g_offs[NGP_L] = {
    0, 4920, 40864, 315496, 2412648, 4509800,
    6606952, 8704104, 10801256, 12898408};

// ---------------------------------------------------------------------
// Kernel A: grid encode + erf-weighted mean over the 64 samples of a ray.
// 1 block = 1 ray, 64 threads = 64 samples. Output feat[B, 40] (L*C).
// ---------------------------------------------------------------------
__global__ __launch_bounds__(64) void ngp_encode_kernel(
    const float* __restrict__ means,   // [B,64,3]
    const float* __restrict__ stds,    // [B,64]
    const float* __restrict__ emb,     // [TOTAL_PARAMS,4]
    float* __restrict__ feat)          // [B,40]
{
  const int b = blockIdx.x;
  const int t = threadIdx.x;           // sample index

  const float* mp = means + ((size_t)b * 64 + t) * 3;
  const float sd  = stds[(size_t)b * 64 + t];

  // encoder: [-1,1] -> [0,1]
  const float x01 = (mp[0] + 1.0f) * 0.5f;
  const float y01 = (mp[1] + 1.0f) * 0.5f;
  const float z01 = (mp[2] + 1.0f) * 0.5f;

  float acc[NGP_L * NGP_C];
#pragma unroll
  for (int i = 0; i < NGP_L * NGP_C; ++i) acc[i] = 0.0f;

  const float inv_sqrt8 = 0.35355339059327373f;  // 1/sqrt(8)

#pragma unroll
  for (int l = 0; l < NGP_L; ++l) {
    const int   res   = 16 << l;                 // grid size used for weight
    const float scale = (float)(res - 1);        // torch-ngp: base*2^l - 1
    const unsigned side = (unsigned)(res + 1);
    const bool hashed = (l >= 3);
    const int  off    = g_offs[l];

    // per-(sample,level) erf weight: erf(1/(sqrt8 * std * res))
    const float wl = erff(inv_sqrt8 / (sd * (float)res));

    const float px = x01 * scale + 0.5f;
    const float py = y01 * scale + 0.5f;
    const float pz = z01 * scale + 0.5f;
    const float gx = floorf(px), gy = floorf(py), gz = floorf(pz);
    const float fx = px - gx, fy = py - gy, fz = pz - gz;
    const unsigned ux = (unsigned)gx, uy = (unsigned)gy, uz = (unsigned)gz;

    const float wx[2] = {1.0f - fx, fx};
    const float wy[2] = {1.0f - fy, fy};
    const float wz[2] = {1.0f - fz, fz};

#pragma unroll
    for (int c = 0; c < 8; ++c) {
      const unsigned ci = (c >> 2) & 1, cj = (c >> 1) & 1, ck = c & 1;
      const unsigned cx = ux + ci, cy = uy + cj, cz = uz + ck;
      unsigned idx;
      if (hashed) {
        idx = (cx ^ (cy * P1) ^ (cz * P2)) & HASH_MASK;
      } else {
        idx = cx + cy * side + cz * side * side;   // always < padded size
      }
      const float4 g =
          *reinterpret_cast<const float4*>(emb + 4 * (size_t)(off + (int)idx));
      const float w = wl * wx[ci] * wy[cj] * wz[ck];
      acc[l * 4 + 0] += w * g.x;
      acc[l * 4 + 1] += w * g.y;
      acc[l * 4 + 2] += w * g.z;
      acc[l * 4 + 3] += w * g.w;
    }
  }

  // reduce the 64 samples of this ray (mean)
  __shared__ float red[64][NGP_L * NGP_C + 1];   // +1 pad breaks bank conflicts
#pragma unroll
  for (int f = 0; f < NGP_L * NGP_C; ++f) red[t][f] = acc[f];
  __syncthreads();

  if (t < NGP_L * NGP_C) {
    float s = 0.0f;
    for (int i = 0; i < 64; ++i) s += red[i][t];
    feat[(size_t)b * (NGP_L * NGP_C) + t] = s * (1.0f / 64.0f);
  }
}

// ---------------------------------------------------------------------
// Kernel B: fused MLP with fp32 WMMA (V_WMMA_F32_16X16X4_F32).
//   h = relu(feat @ w1 + b1)   [8192,40]@[40,64]
//   x = h @ w2 + b2            [8192,64]@[64,256]
//   density = softplus(x[:,0] - 1)
// Block: 128 threads = 4 waves, covers 16 rows.
// Fragment layouts per ISA 7.12.2 (32-bit A 16x4 / 32-bit C 16x16):
//   A: lane m = lane&15; v0 holds K = k0+(lane<16?0:2), v1 holds K+1.
//   B: lane n = lane&15; v0 holds K = k0+(lane<16?0:2), v1 holds K+1.
//   C/D: vgpr r -> row = r + (lane<16?0:8), col = lane&15.
// ---------------------------------------------------------------------
typedef __attribute__((ext_vector_type(2))) float v2f;
typedef __attribute__((ext_vector_type(8))) float v8f;

__global__ __launch_bounds__(128) void ngp_mlp_kernel(
    const float* __restrict__ feat,   // [B,40]
    const float* __restrict__ w1,     // [40,64]
    const float* __restrict__ b1,     // [64]
    const float* __restrict__ w2,     // [64,256]
    const float* __restrict__ b2,     // [256]
    float* __restrict__ out_density,  // [B]
    float* __restrict__ out_x)        // [B,256]
{
  __shared__ float F[16][41];   // feat tile, padded (gcd(41,64)=1)
  __shared__ float H[16][65];   // hidden tile, padded

  const int b0   = blockIdx.x * 16;
  const int tid  = threadIdx.x;
  const int wave = tid >> 5;
  const int lane = tid & 31;
  const bool hi  = lane >= 16;
  const int  lm  = lane & 15;

  // stage feat tile [16 x 40] into LDS
  for (int i = tid; i < 16 * 40; i += 128)
    F[i / 40][i % 40] = feat[(size_t)b0 * 40 + i];
  __syncthreads();

  // ---- stage 1: h-tile, wave w owns columns 16w..16w+15 ----
  {
    const int n0 = wave * 16;
    v8f c;
    const float bias = b1[n0 + lm];
#pragma unroll
    for (int r = 0; r < 8; ++r) c[r] = bias;

#pragma unroll
    for (int kk = 0; kk < 10; ++kk) {
      const int ka = kk * 4 + (hi ? 2 : 0);
      v2f a, bb;
      a.x  = F[lm][ka];
      a.y  = F[lm][ka + 1];
      bb.x = w1[ka * 64 + n0 + lm];
      bb.y = w1[(ka + 1) * 64 + n0 + lm];
      c = __builtin_amdgcn_wmma_f32_16x16x4_f32(false, a, false, bb,
                                                (short)0, c, false, false);
    }
#pragma unroll
    for (int r = 0; r < 8; ++r) {
      const float v = c[r] > 0.0f ? c[r] : 0.0f;         // ReLU
      H[r + (hi ? 8 : 0)][n0 + lm] = v;
    }
  }
  __syncthreads();

  // ---- stage 2: 16 N-tiles of 16; wave w does tiles w, w+4, w+8, w+12 ----
  for (int tt = 0; tt < 4; ++tt) {
    const int n0 = (wave + tt * 4) * 16;
    v8f c;
    const float bias = b2[n0 + lm];
#pragma unroll
    for (int r = 0; r < 8; ++r) c[r] = bias;

#pragma unroll
    for (int kk = 0; kk < 16; ++kk) {
      const int ka = kk * 4 + (hi ? 2 : 0);
      v2f a, bb;
      a.x  = H[lm][ka];
      a.y  = H[lm][ka + 1];
      bb.x = w2[ka * 256 + n0 + lm];
      bb.y = w2[(ka + 1) * 256 + n0 + lm];
      c = __builtin_amdgcn_wmma_f32_16x16x4_f32(false, a, false, bb,
                                                (short)0, c, false, false);
    }
#pragma unroll
    for (int r = 0; r < 8; ++r) {
      const int row = b0 + r + (hi ? 8 : 0);
      const int col = n0 + lm;
      const float v = c[r];
      out_x[(size_t)row * 256 + col] = v;
      if (col == 0) {
        // softplus(v - 1), numerically stable: max(z,0) + log1p(exp(-|z|))
        const float z = v - 1.0f;
        out_density[row] = fmaxf(z, 0.0f) + log1pf(expf(-fabsf(z)));
      }
    }
  }
}

// ---------------------------------------------------------------------
extern "C" void kernel_launch(void* const* d_in, const int* in_sizes, int n_in,
                              void* d_out, int out_size, void* d_ws, size_t ws_size,
                              hipStream_t stream) {
  const float* means = (const float*)d_in[0];   // [8192,64,3]
  const float* stds  = (const float*)d_in[1];   // [8192,64]
  const float* emb   = (const float*)d_in[2];   // [14995560,4]
  const float* w1    = (const float*)d_in[3];   // [40,64]
  const float* b1    = (const float*)d_in[4];   // [64]
  const float* w2    = (const float*)d_in[5];   // [64,256]
  const float* b2    = (const float*)d_in[6];   // [256]

  const int B = in_sizes[1] / 64;               // 8192 rays

  float* feat        = (float*)d_ws;            // [B,40] scratch
  float* out_density = (float*)d_out;           // [B]
  float* out_x       = (float*)d_out + B;       // [B,256]

  ngp_encode_kernel<<<B, 64, 0, stream>>>(means, stds, emb, feat);
  ngp_mlp_kernel<<<B / 16, 128, 0, stream>>>(feat, w1, b1, w2, b2,
                                             out_density, out_x);
}